// MembranePotentialDecoder_68401649156453
// MI455X (gfx1250) — compile-verified
//
#include <hip/hip_runtime.h>
#include <hip/hip_bf16.h>
#include <stdint.h>

// Problem constants (from reference): spikes (B,T,N) f32, out (B,N) f32
#define B_DIM   32
#define T_DIM   512
#define N_DIM   2048
#define TAU_F   10.0f
#define DECAY_F 0.90483741803595952f   // exp(-1/10)

#define CHUNKS_MAX 16                  // T split for parallelism
#define DEPTH      8                   // timesteps per async batch (double buffered)
#define THREADS    256                 // 8 waves / block
#define NT         (THREADS * 4)       // 1024 floats of N per block
#define NTILES     (N_DIM / NT)        // 2

typedef int v4i __attribute__((ext_vector_type(4)));
typedef __attribute__((address_space(1))) v4i* g_v4i_p;   // global v4i32*
typedef __attribute__((address_space(3))) v4i* l_v4i_p;   // LDS    v4i32*

// --- CDNA5 async global->LDS b128 (per-lane 16B, tracked by ASYNCcnt) ------
static __device__ __forceinline__ void async_ld_b128(const void* g, void* l) {
#if __has_builtin(__builtin_amdgcn_global_load_async_to_lds_b128)
  g_v4i_p gp = (g_v4i_p)const_cast<void*>(g);
  l_v4i_p lp = (l_v4i_p)l;
  __builtin_amdgcn_global_load_async_to_lds_b128(gp, lp, /*offset=*/0, /*cpol=*/0);
#else
  uint32_t loff = (uint32_t)(size_t)(l_v4i_p)l;   // 32-bit LDS byte address
  uint64_t ga   = (uint64_t)(size_t)g;
  asm volatile("global_load_async_to_lds_b128 %0, %1, off"
               :: "v"(loff), "v"(ga) : "memory");
#endif
}

// s_wait_asynccnt <= n : DMA-complete gate for the oldest in-flight batch.
#define WAIT_ASYNC(n) asm volatile("s_wait_asynccnt %0" :: "n"(n) : "memory")
// s_wait_dscnt 0 : all ds_loads retired -> safe for async DMA to overwrite LDS.
#define WAIT_DS0()    asm volatile("s_wait_dscnt 0x0" ::: "memory")

// ---------------------------------------------------------------------------
// Kernel 1: one block = (batch b, time-chunk c, n-tile). Streams its chunk of
// the time axis through an LDS double buffer via async DMA and accumulates
// acc = acc*decay + s  (forward order => weight decay^(steps-1-i) per chunk).
// Each lane consumes only the 16B it DMA'd itself -> no workgroup barriers;
// only per-wave ASYNCcnt / DScnt gating.
// ---------------------------------------------------------------------------
template <int CHUNKS>
__global__ __launch_bounds__(THREADS)
void MembranePotentialDecoder_partial_kernel(const float* __restrict__ spikes,
                                             float* __restrict__ dst) {
  __shared__ float4 stage[2][DEPTH][THREADS];   // 2*8*256*16B = 64 KB

  constexpr int STEPS  = T_DIM / CHUNKS;        // timesteps in this chunk
  constexpr int NBATCH = STEPS / DEPTH;         // >= 2 for CHUNKS in {1,16}

  const int tid  = threadIdx.x;
  int blk        = blockIdx.x;
  const int tile = blk % NTILES;  blk /= NTILES;   // power-of-2: mask/shift
  const int c    = blk % CHUNKS;  blk /= CHUNKS;   // power-of-2: mask/shift
  const int b    = blk;

  const int n0    = tile * NT + tid * 4;
  const int t0    = c * STEPS;
  const float* gp = spikes + (((size_t)b * T_DIM + t0) * N_DIM + n0);

  // Prologue: prime both buffers (2*DEPTH async ops in flight per wave)
  #pragma unroll
  for (int p = 0; p < 2; ++p) {
    #pragma unroll
    for (int d = 0; d < DEPTH; ++d) {
      const int t = p * DEPTH + d;
      async_ld_b128(gp + (size_t)t * N_DIM, &stage[p][d][tid]);
    }
  }

  float4 acc = {0.f, 0.f, 0.f, 0.f};
  int buf = 0;
  for (int k = 0; k < NBATCH; ++k) {
    if (k + 1 < NBATCH) { WAIT_ASYNC(DEPTH); }  // oldest batch landed
    else                { WAIT_ASYNC(0);     }  // last batch: drain all

    #pragma unroll
    for (int d = 0; d < DEPTH; ++d) {
      float4 s = stage[buf][d][tid];            // ds_load_b128
      acc.x = acc.x * DECAY_F + s.x;
      acc.y = acc.y * DECAY_F + s.y;
      acc.z = acc.z * DECAY_F + s.z;
      acc.w = acc.w * DECAY_F + s.w;
    }

    const int kn = k + 2;                       // refill the buffer just freed
    if (kn < NBATCH) {
      WAIT_DS0();   // ds reads retired before DMA may overwrite this buffer
      #pragma unroll
      for (int d = 0; d < DEPTH; ++d) {
        const int t = kn * DEPTH + d;
        async_ld_b128(gp + (size_t)t * N_DIM, &stage[buf][d][tid]);
      }
    }
    buf ^= 1;
  }

  // dst layout: [b][c][n]; for CHUNKS==1 this aliases the (b,n) output layout.
  *(float4*)&dst[((size_t)b * CHUNKS + c) * N_DIM + n0] = acc;
}

// ---------------------------------------------------------------------------
// Kernel 2: combine chunk partials: out = sum_c partial_c * decay^(STEPS*(C-1-c))
// ---------------------------------------------------------------------------
template <int CHUNKS>
__global__ __launch_bounds__(256)
void MembranePotentialDecoder_combine_kernel(const float* __restrict__ part,
                                             float* __restrict__ out) {
  constexpr int STEPS = T_DIM / CHUNKS;
  const int idx = blockIdx.x * blockDim.x + threadIdx.x;  // one per (b, n/4)
  const int b   = idx / (N_DIM / 4);
  const int n0  = (idx % (N_DIM / 4)) * 4;

  float4 acc = {0.f, 0.f, 0.f, 0.f};
  #pragma unroll
  for (int c = 0; c < CHUNKS; ++c) {
    const float w = __expf(-(float)(STEPS * (CHUNKS - 1 - c)) / TAU_F);
    float4 p = *(const float4*)&part[((size_t)b * CHUNKS + c) * N_DIM + n0];
    acc.x += w * p.x;
    acc.y += w * p.y;
    acc.z += w * p.z;
    acc.w += w * p.w;
  }
  *(float4*)&out[(size_t)b * N_DIM + n0] = acc;
}

// ---------------------------------------------------------------------------
extern "C" void kernel_launch(void* const* d_in, const int* in_sizes, int n_in,
                              void* d_out, int out_size, void* d_ws, size_t ws_size,
                              hipStream_t stream) {
  (void)in_sizes; (void)n_in; (void)out_size;
  const float* spikes = (const float*)d_in[0];
  float*       out    = (float*)d_out;

  const size_t need = (size_t)B_DIM * CHUNKS_MAX * N_DIM * sizeof(float); // 4 MB
  const bool chunked = (d_ws != nullptr && ws_size >= need);

  if (chunked) {
    const int grid1 = B_DIM * CHUNKS_MAX * NTILES;   // 1024 blocks
    MembranePotentialDecoder_partial_kernel<CHUNKS_MAX>
        <<<grid1, THREADS, 0, stream>>>(spikes, (float*)d_ws);
    const int total = B_DIM * (N_DIM / 4);           // 16384 threads
    MembranePotentialDecoder_combine_kernel<CHUNKS_MAX>
        <<<total / 256, 256, 0, stream>>>((const float*)d_ws, out);
  } else {
    const int grid1 = B_DIM * 1 * NTILES;            // 64 blocks (fallback)
    MembranePotentialDecoder_partial_kernel<1>
        <<<grid1, THREADS, 0, stream>>>(spikes, out);
  }
}